// SmoothSTEMinMax_35905926595272
// MI455X (gfx1250) — compile-verified
//
#include <hip/hip_runtime.h>
#include <stdint.h>

// Problem sizes (fixed by the reference).
constexpr int B = 512;      // batch rows of x
constexpr int I = 512;      // reduction dim
constexpr int O = 1024;     // output cols
// Tiling.
constexpr int BM = 64;      // b-tile per workgroup
constexpr int BN = 64;      // o-tile per workgroup
constexpr int KT = 32;      // k-tile
constexpr int SPLITK = 4;   // K slices -> 512 workgroups for occupancy
constexpr int KS = I / SPLITK;   // 128 k per slice
constexpr int NT = KS / KT;      // 4 k-tiles per slice
constexpr int LDW = 68;     // padded LDS row stride in floats (272B, multiple of 16B)

typedef __attribute__((ext_vector_type(4))) float v4f;

__device__ __forceinline__ unsigned lds_off(const void* p) {
  // Generic pointer to __shared__ carries the LDS byte offset in its low 32 bits.
  return (unsigned)(unsigned long long)p;
}

__device__ __forceinline__ void async_copy_b32(unsigned lds, unsigned long long gaddr) {
  asm volatile("global_load_async_to_lds_b32 %0, %1, off"
               :: "v"(lds), "v"(gaddr) : "memory");
}
__device__ __forceinline__ void async_copy_b128(unsigned lds, unsigned long long gaddr) {
  asm volatile("global_load_async_to_lds_b128 %0, %1, off"
               :: "v"(lds), "v"(gaddr) : "memory");
}

// Seed output with +INF (identity of min) so split-K slices can atomically combine.
__global__ __launch_bounds__(256)
void InitInf_kernel(float* __restrict__ out) {
  unsigned long long idx = (unsigned long long)blockIdx.x * 256 + threadIdx.x;
  v4f v = {__builtin_inff(), __builtin_inff(), __builtin_inff(), __builtin_inff()};
  *(v4f*)&out[idx * 4] = v;
}

__global__ __launch_bounds__(256)
void SmoothSTEMinMax_kernel(const float* __restrict__ x,
                            const float* __restrict__ w,
                            float* __restrict__ out) {
  // Double-buffered tiles. sX holds the x tile TRANSPOSED: sX[k][b].
  __shared__ __attribute__((aligned(16))) float sX[2][KT][LDW];
  __shared__ __attribute__((aligned(16))) float sW[2][KT][LDW]; // sW[k][o]

  const int tid = threadIdx.x;
  const int o0  = blockIdx.x * BN;
  const int b0  = blockIdx.y * BM;
  const int kz  = blockIdx.z * KS;   // this slice's K origin
  const int tx  = tid & 15;          // o-direction thread coord (x4 floats)
  const int ty  = tid >> 4;          // b-direction thread coord (x4 floats)

  // Issue the async DMA for k-tile `kt` of this slice into buffer `buf`:
  // 10 async ops per thread, tracked by ASYNCcnt.
  auto issue = [&](int kt, int buf) {
    const int k0 = kz + kt * KT;
    // x tile: 64x32 elements, transposed scatter into LDS. 8 b32 ops per thread.
#pragma unroll
    for (int e = 0; e < 8; ++e) {
      int idx = tid + e * 256;              // 0..2047
      int k   = idx & (KT - 1);
      int b   = idx >> 5;                   // 0..63
      unsigned l = lds_off(&sX[buf][k][b]);
      unsigned long long g =
          (unsigned long long)(const void*)(x + (unsigned long long)(b0 + b) * I + (k0 + k));
      async_copy_b32(l, g);
    }
    // w tile: 32 rows x 64 floats, contiguous 16B chunks. 2 b128 ops per thread.
#pragma unroll
    for (int e = 0; e < 2; ++e) {
      int idx  = tid + e * 256;             // 0..511
      int lane = idx & 15;                  // 16B chunk within row
      int kk   = idx >> 4;                  // 0..31
      unsigned l = lds_off(&sW[buf][kk][lane * 4]);
      unsigned long long g =
          (unsigned long long)(const void*)(w + (unsigned long long)(k0 + kk) * O + o0 + lane * 4);
      async_copy_b128(l, g);
    }
  };

  float acc[4][4];
#pragma unroll
  for (int i = 0; i < 4; ++i)
#pragma unroll
    for (int j = 0; j < 4; ++j)
      acc[i][j] = __builtin_inff();         // identity of min

  issue(0, 0);                              // prologue: DMA tile 0

  for (int t = 0; t < NT; ++t) {
    if (t + 1 < NT) {
      issue(t + 1, (t + 1) & 1);            // overlap next tile's DMA with compute
      asm volatile("s_wait_asynccnt 10" ::: "memory");  // retire tile t's 10 ops
    } else {
      asm volatile("s_wait_asynccnt 0" ::: "memory");
    }
    __syncthreads();                        // tile t fully in LDS for all 8 waves

    const float (*Xs)[LDW] = sX[t & 1];
    const float (*Ws)[LDW] = sW[t & 1];
#pragma unroll 8
    for (int k = 0; k < KT; ++k) {
      v4f xv = *(const v4f*)&Xs[k][ty * 4]; // ds_load_b128 (broadcast across tx)
      v4f wv = *(const v4f*)&Ws[k][tx * 4]; // ds_load_b128 (conflict-free across tx)
#pragma unroll
      for (int i = 0; i < 4; ++i)
#pragma unroll
        for (int j = 0; j < 4; ++j) {
          // Fused tropical FMA: acc = min(max(x, w), acc) in ONE VALU op.
          asm("v_maxmin_num_f32 %0, %1, %2, %0"
              : "+v"(acc[i][j])
              : "v"(xv[i]), "v"(wv[j]));
        }
    }
    __syncthreads();                        // protect this buffer from next issue()
  }

  // Combine split-K slices with native fp32 min atomics (exact & order-independent).
#pragma unroll
  for (int i = 0; i < 4; ++i) {
#pragma unroll
    for (int j = 0; j < 4; ++j) {
      unsigned long long g = (unsigned long long)(const void*)
          &out[(unsigned long long)(b0 + ty * 4 + i) * O + (o0 + tx * 4 + j)];
      asm volatile("global_atomic_min_num_f32 %0, %1, off"
                   :: "v"(g), "v"(acc[i][j]) : "memory");
    }
  }
}

extern "C" void kernel_launch(void* const* d_in, const int* in_sizes, int n_in,
                              void* d_out, int out_size, void* d_ws, size_t ws_size,
                              hipStream_t stream) {
  (void)in_sizes; (void)n_in; (void)d_ws; (void)ws_size; (void)out_size;
  const float* x = (const float*)d_in[0];   // [512, 512]
  const float* w = (const float*)d_in[1];   // [512, 1024]
  float* out = (float*)d_out;               // [512, 1024]

  // Seed d_out = +INF every call (harness poisons buffers; launch must be
  // deterministic and self-contained).
  InitInf_kernel<<<(B * O / 4) / 256, 256, 0, stream>>>(out);

  dim3 grid(O / BN, B / BM, SPLITK);        // (16, 8, 4) = 512 workgroups
  SmoothSTEMinMax_kernel<<<grid, 256, 0, stream>>>(x, w, out);
}